// SparseDIA_76501957477134
// MI455X (gfx1250) — compile-verified
//
#include <hip/hip_runtime.h>
#include <cstdint>
#include <cstddef>

typedef float v4f __attribute__((ext_vector_type(4)));

// Problem constants (match reference).
constexpr int N = 8192;            // rows of A / other
constexpr int M = 4096;            // dense columns
constexpr int NOFF = 9;
constexpr int OFFS[NOFF] = {-128, -64, -8, -1, 0, 1, 8, 64, 128};

// Tiling.
constexpr int TILE_COLS      = 32;   // floats per block column tile (128 B rows)
constexpr int ROWS_PER_BLOCK = 128;  // output rows per block
constexpr int HALO           = 128;  // max |offset|
constexpr int IN_ROWS        = ROWS_PER_BLOCK + 2 * HALO;  // 384 staged rows
constexpr int THREADS        = 256;
constexpr int CHUNKS_PER_THREAD = (IN_ROWS * TILE_COLS / 4) / THREADS; // 12 x b128

__global__ __launch_bounds__(THREADS)
void dia_spmm_kernel(const float* __restrict__ diags,
                     const float* __restrict__ other,
                     float* __restrict__ out) {
    __shared__ alignas(16) float tile[IN_ROWS * TILE_COLS];      // 48 KB
    __shared__ float coef[NOFF * ROWS_PER_BLOCK];                // 4.5 KB

    const int tid = threadIdx.x;
    const int c0  = blockIdx.x * TILE_COLS;       // column tile start
    const int i0  = blockIdx.y * ROWS_PER_BLOCK;  // output row block start

    const uint32_t tile_lds = (uint32_t)(uintptr_t)(&tile[0]);

    // ---- Stage input rows [i0-HALO, i0+ROWS_PER_BLOCK+HALO) via CDNA5 async
    // copies (global -> LDS, ASYNCcnt-tracked). Row index clamped to [0,N);
    // out-of-range rows are neutralized by zeroed coefficients below.
    #pragma unroll
    for (int t = 0; t < CHUNKS_PER_THREAD; ++t) {
        const int q  = t * THREADS + tid;
        const int lr = q >> 3;              // staged row 0..383
        const int cc = (q & 7) << 2;        // float col offset 0,4,...,28
        int r = i0 - HALO + lr;
        r = r < 0 ? 0 : (r >= N ? N - 1 : r);
        const float* gp = other + (size_t)r * M + c0 + cc;
        const uint32_t lp = tile_lds + (uint32_t)((lr * TILE_COLS + cc) * 4);
        asm volatile("global_load_async_to_lds_b128 %0, %1, off"
                     :: "v"(lp), "v"(gp)
                     : "memory");
    }

    // ---- Masked diagonal coefficients -> LDS (overlaps with async copies).
    // coef[k][lr] = diags[k, i0+lr+off_k] if row in range else 0.
    if (tid < ROWS_PER_BLOCK) {
        #pragma unroll
        for (int k = 0; k < NOFF; ++k) {
            const int r = i0 + tid + OFFS[k];
            coef[k * ROWS_PER_BLOCK + tid] =
                (r >= 0 && r < N) ? diags[k * N + r] : 0.0f;
        }
    }

    // Wait for this wave's async LDS writes, then cross-wave barrier.
    asm volatile("s_wait_asynccnt 0x0" ::: "memory");
    __syncthreads();

    // ---- Compute: each thread produces 4 float4 outputs.
    const int col = (tid & 7) << 2;   // 0,4,...,28
    const int rb  = tid >> 3;         // 0..31
    #pragma unroll
    for (int s = 0; s < 4; ++s) {
        const int lr = rb + 32 * s;   // local output row 0..127
        v4f acc = {0.f, 0.f, 0.f, 0.f};
        #pragma unroll
        for (int k = 0; k < NOFF; ++k) {
            const int lrin = lr + HALO + OFFS[k];       // 0..383, in range
            const v4f v =
                *reinterpret_cast<const v4f*>(&tile[lrin * TILE_COLS + col]);
            const float c = coef[k * ROWS_PER_BLOCK + lr];
            acc += c * v;   // vector FMA
        }
        // Output is write-once: stream it past the caches (keep L2 for `other`).
        v4f* op = reinterpret_cast<v4f*>(
            out + (size_t)(i0 + lr) * M + c0 + col);
        __builtin_nontemporal_store(acc, op);
    }
}

extern "C" void kernel_launch(void* const* d_in, const int* in_sizes, int n_in,
                              void* d_out, int out_size, void* d_ws, size_t ws_size,
                              hipStream_t stream) {
    (void)in_sizes; (void)n_in; (void)out_size; (void)d_ws; (void)ws_size;
    const float* diags = (const float*)d_in[0];   // (9, N) fp32
    const float* other = (const float*)d_in[1];   // (N, M) fp32
    float* out = (float*)d_out;                   // (N, M) fp32

    dim3 grid(M / TILE_COLS, N / ROWS_PER_BLOCK); // (128, 64)
    dim3 block(THREADS);
    dia_spmm_kernel<<<grid, block, 0, stream>>>(diags, other, out);
}